// StochasticPool2DLayer_27315992003274
// MI455X (gfx1250) — compile-verified
//
#include <hip/hip_runtime.h>
#include <hip/hip_bf16.h>
#include <stdint.h>

#define AS1 __attribute__((address_space(1)))
#define AS3 __attribute__((address_space(3)))

typedef int v4i __attribute__((ext_vector_type(4)));

#if defined(__has_builtin)
#if __has_builtin(__builtin_amdgcn_global_load_async_to_lds_b128)
#define HAVE_ASYNC_LDS 1
#endif
#if __has_builtin(__builtin_amdgcn_s_wait_asynccnt)
#define HAVE_WAIT_ASYNC 1
#endif
#endif

// ---------------------------------------------------------------------------
// Threefry-2x32 (20 rounds), bit-exact with jax._src.prng.threefry2x32
// ---------------------------------------------------------------------------
__device__ __forceinline__ void tf_round(unsigned &x0, unsigned &x1, int r) {
  x0 += x1;
  x1 = (x1 << r) | (x1 >> (32 - r));
  x1 ^= x0;
}

__device__ __forceinline__ void threefry2x32(unsigned k0, unsigned k1,
                                             unsigned c0, unsigned c1,
                                             unsigned &o0, unsigned &o1) {
  const unsigned ks2 = k0 ^ k1 ^ 0x1BD11BDAu;
  unsigned x0 = c0 + k0;
  unsigned x1 = c1 + k1;
  const int ra[4] = {13, 15, 26, 6};
  const int rb[4] = {17, 29, 16, 24};
#pragma unroll
  for (int i = 0; i < 4; ++i) tf_round(x0, x1, ra[i]);
  x0 += k1;  x1 += ks2 + 1u;
#pragma unroll
  for (int i = 0; i < 4; ++i) tf_round(x0, x1, rb[i]);
  x0 += ks2; x1 += k0 + 2u;
#pragma unroll
  for (int i = 0; i < 4; ++i) tf_round(x0, x1, ra[i]);
  x0 += k0;  x1 += k1 + 3u;
#pragma unroll
  for (int i = 0; i < 4; ++i) tf_round(x0, x1, rb[i]);
  x0 += k1;  x1 += ks2 + 4u;
#pragma unroll
  for (int i = 0; i < 4; ++i) tf_round(x0, x1, ra[i]);
  x0 += ks2; x1 += k0 + 5u;
  o0 = x0;
  o1 = x1;
}

// ---------------------------------------------------------------------------
// Kernel 1: reproduce jax.random key(123) -> split -> uniform(64,2) -> argsort
// idx[0..63]   = idx_w  (row picks, one per 2-wide cell)
// idx[64..127] = idx_h  (col picks)
// ---------------------------------------------------------------------------
__global__ __launch_bounds__(64) void stochpool_idx_kernel(int *__restrict__ idx) {
  __shared__ unsigned bw[128];
  __shared__ unsigned bh[128];
  const int t = threadIdx.x;  // 0..63

  // key = threefry_seed(123) = (0, 123); split(key, 2):
  // counts = [0,1,2,3]; block0 encrypts (0,2), block1 encrypts (1,3)
  // kw = (o0(b0), o0(b1)); kh = (o1(b0), o1(b1))
  unsigned a0, a1, b0, b1;
  threefry2x32(0u, 123u, 0u, 2u, a0, a1);
  threefry2x32(0u, 123u, 1u, 3u, b0, b1);
  const unsigned kw0 = a0, kw1 = b0;
  const unsigned kh0 = a1, kh1 = b1;

  // uniform(key, (64,2)): bits = threefry_2x32(key, iota(128))
  // block t encrypts (t, t+64) -> bits[t], bits[t+64]
  unsigned o0, o1;
  threefry2x32(kw0, kw1, (unsigned)t, (unsigned)t + 64u, o0, o1);
  bw[t] = o0;
  bw[t + 64] = o1;
  threefry2x32(kh0, kh1, (unsigned)t, (unsigned)t + 64u, o0, o1);
  bh[t] = o0;
  bh[t + 64] = o1;
  __syncthreads();

  // u[c,g] = float((bits[2c+g] >> 9) | 0x3f800000) - 1  -> monotone in bits>>9
  // stable argsort of 2 -> pick index of min, ties -> 0
  {
    const unsigned u0 = bw[2 * t] >> 9, u1 = bw[2 * t + 1] >> 9;
    idx[t] = 2 * t + ((u0 <= u1) ? 0 : 1);
  }
  {
    const unsigned u0 = bh[2 * t] >> 9, u1 = bh[2 * t + 1] >> 9;
    idx[64 + t] = 2 * t + ((u0 <= u1) ? 0 : 1);
  }
}

// ---------------------------------------------------------------------------
// Kernel 2: per (n,c) plane -- async-stage 128x128 f32 plane into LDS (64 KB),
// then gather 2x2 max windows at (idx_w[i], idx_h[j]).
// out[i][j] = 0 if idx==127 (zero pad of the 127x127 VALID pool result).
// ---------------------------------------------------------------------------
__global__ __launch_bounds__(256) void stochpool_main_kernel(
    const float *__restrict__ x, const int *__restrict__ idx,
    float *__restrict__ out) {
  __shared__ float plane[128 * 128];  // 64 KB of the WGP's 320 KB LDS
  __shared__ int iw[64];              // staged row picks (broadcast reads)
  const int t = threadIdx.x;          // 0..255
  const int p = blockIdx.x;           // plane = n*256 + c
  const float *src = x + (size_t)p * (128 * 128);

  if (t < 64) iw[t] = idx[t];

  // Column pick is loop-invariant per thread: j = (k*256 + t) & 63 = t & 63.
  const int h = idx[64 + (t & 63)];

#if defined(HAVE_ASYNC_LDS)
  // 4096 16-byte chunks per plane; 16 async b128 copies per thread.
#pragma unroll
  for (int q = 0; q < 16; ++q) {
    const int e = q * 256 + t;  // chunk index
    __builtin_amdgcn_global_load_async_to_lds_b128(
        (AS1 v4i *)(src + e * 4), (AS3 v4i *)(&plane[e * 4]), 0, 0);
  }
#if defined(HAVE_WAIT_ASYNC)
  __builtin_amdgcn_s_wait_asynccnt(0);
#else
  asm volatile("s_wait_asynccnt 0" ::: "memory");
#endif
#else
  // Fallback: coalesced b128 loads through VGPRs + ds stores.
#pragma unroll
  for (int q = 0; q < 16; ++q) {
    const int e = q * 256 + t;
    ((float4 *)plane)[e] = ((const float4 *)src)[e];
  }
#endif
  __syncthreads();

  float *dst = out + (size_t)p * (64 * 64) + t;
  const int i0 = t >> 6;       // 0..3: row offset of this thread's quarter
  const bool hok = (h < 127);
#pragma unroll
  for (int k = 0; k < 16; ++k) {
    const int i = 4 * k + i0;  // output row 0..63
    const int w = iw[i];       // sampled row in padded pooled map
    float r = 0.0f;
    if (w < 127 && hok) {
      const float *r0 = &plane[w * 128 + h];
      const float *r1 = r0 + 128;
      r = fmaxf(fmaxf(r0[0], r0[1]), fmaxf(r1[0], r1[1]));
    }
    dst[k * 256] = r;  // lane-contiguous -> coalesced b32 stores
  }
}

// ---------------------------------------------------------------------------
extern "C" void kernel_launch(void *const *d_in, const int *in_sizes, int n_in,
                              void *d_out, int out_size, void *d_ws,
                              size_t ws_size, hipStream_t stream) {
  const float *x = (const float *)d_in[0];
  float *out = (float *)d_out;
  int *idx = (int *)d_ws;  // needs 128 ints = 512 B

  const int planes = in_sizes[0] / (128 * 128);  // 16 * 256 = 4096

  stochpool_idx_kernel<<<1, 64, 0, stream>>>(idx);
  stochpool_main_kernel<<<planes, 256, 0, stream>>>(x, idx, out);
  (void)n_in;
  (void)out_size;
  (void)ws_size;
}